// LSTM_16063177687549
// MI455X (gfx1250) — compile-verified
//
#include <hip/hip_runtime.h>

typedef __attribute__((ext_vector_type(2))) float v2f;
typedef __attribute__((ext_vector_type(8))) float v8f;

#define LB 8192
#define LT 2048
#define LI 2
#define LH 2

__device__ __forceinline__ float fast_sigmoid(float x) {
    // 1 / (1 + e^-x) = rcp(1 + exp2(-x * log2e))
    float e = __builtin_amdgcn_exp2f(x * -1.4426950408889634f);
    return __builtin_amdgcn_rcpf(1.0f + e);
}

__device__ __forceinline__ float fast_tanh(float x) {
    // tanh(x) = 1 - 2/(e^{2x} + 1),  e^{2x} = exp2(x * 2*log2e)
    float e = __builtin_amdgcn_exp2f(x * 2.8853900817779268f);
    return 1.0f - 2.0f * __builtin_amdgcn_rcpf(e + 1.0f);
}

__global__ __launch_bounds__(256) void lstm_wmma_kernel(
    const float* __restrict__ x,     // [B, T, I]
    const float* __restrict__ W_ih,  // [8, 2] row-major, gate order i,f,g,o
    const float* __restrict__ W_hh,  // [8, 2]
    const float* __restrict__ b_ih,  // [8]
    const float* __restrict__ b_hh,  // [8]
    const float* __restrict__ W_fc,  // [1, 2]
    const float* __restrict__ b_fc,  // [1]
    float* __restrict__ out)         // [B]
{
    const int lane = threadIdx.x & 31;
    const int wave = (blockIdx.x * blockDim.x + threadIdx.x) >> 5;
    const int n    = lane & 15;        // batch column within this wave's group
    const bool hi  = lane >= 16;       // high half carries the K=2..3 (h / W_hh) slice
    const int batch = wave * 16 + n;

    // ---- A matrix (constant): 16x4 f32, M = gate row (rows 8-15 duplicate 0-7)
    //   lane<16 : A[M][0..1] = W_ih[M&7][:]   (vgpr0 = K0, vgpr1 = K1)
    //   lane>=16: A[M][2..3] = W_hh[M&7][:]   (vgpr0 = K2, vgpr1 = K3)
    const int g = n & 7;
    v2f a;
    a[0] = hi ? W_hh[g * 2 + 0] : W_ih[g * 2 + 0];
    a[1] = hi ? W_hh[g * 2 + 1] : W_ih[g * 2 + 1];

    // ---- C matrix: bias broadcast. D[vgpr r] = gate r (rows duplicated), so
    // every lane's C element r is bias[r].
    v8f cbias;
#pragma unroll
    for (int r = 0; r < 8; ++r) cbias[r] = b_ih[r] + b_hh[r];

    const float4* xrow = (const float4*)(x + (size_t)batch * (LT * LI));

    float h0 = 0.0f, h1 = 0.0f, c0 = 0.0f, c1 = 0.0f;

#pragma unroll 2
    for (int t2 = 0; t2 < LT / 2; ++t2) {
        // 2 timesteps of this lane's batch row: {x[t][0], x[t][1], x[t+1][0], x[t+1][1]}
        float4 xv = xrow[t2];
        if (t2 + 16 < LT / 2) __builtin_prefetch(&xrow[t2 + 16], 0, 0);

#pragma unroll
        for (int s = 0; s < 2; ++s) {
            const float xa = s ? xv.z : xv.x;
            const float xb = s ? xv.w : xv.y;

            // B matrix: 4x16 f32, N = batch column (lane&15)
            //   lane<16 : B[0..1][n] = x_t   lane>=16: B[2..3][n] = h_t
            v2f b;
            b[0] = hi ? h0 : xa;
            b[1] = hi ? h1 : xb;

            // gates[16x8] = [x|h] * [W_ih^T ; W_hh^T] + bias  — one WMMA
            v8f d = __builtin_amdgcn_wmma_f32_16x16x4_f32(
                false, a, false, b, (short)0, cbias, false, false);

            // d[r] = gate r for batch (lane&15), identical in both lane halves.
            const float ig0 = fast_sigmoid(d[0]);
            const float ig1 = fast_sigmoid(d[1]);
            const float fg0 = fast_sigmoid(d[2]);
            const float fg1 = fast_sigmoid(d[3]);
            const float gg0 = fast_tanh(d[4]);
            const float gg1 = fast_tanh(d[5]);
            const float og0 = fast_sigmoid(d[6]);
            const float og1 = fast_sigmoid(d[7]);

            c0 = fg0 * c0 + ig0 * gg0;
            c1 = fg1 * c1 + ig1 * gg1;
            h0 = og0 * fast_tanh(c0);
            h1 = og1 * fast_tanh(c1);
        }
    }

    // fc on final hidden state; every batch's h is valid in lanes 0..15.
    if (!hi) {
        out[batch] = h0 * W_fc[0] + h1 * W_fc[1] + b_fc[0];
    }
}

extern "C" void kernel_launch(void* const* d_in, const int* in_sizes, int n_in,
                              void* d_out, int out_size, void* d_ws, size_t ws_size,
                              hipStream_t stream) {
    (void)in_sizes; (void)n_in; (void)d_ws; (void)ws_size; (void)out_size;
    const float* x    = (const float*)d_in[0];
    const float* W_ih = (const float*)d_in[1];
    const float* W_hh = (const float*)d_in[2];
    const float* b_ih = (const float*)d_in[3];
    const float* b_hh = (const float*)d_in[4];
    const float* W_fc = (const float*)d_in[5];
    const float* b_fc = (const float*)d_in[6];
    float* out = (float*)d_out;

    // 8192 batches / 16 per wave = 512 waves; 256 threads (8 waves) per block.
    const int threads = 256;
    const int waves   = LB / 16;                 // 512
    const int blocks  = waves * 32 / threads;    // 64

    lstm_wmma_kernel<<<blocks, threads, 0, stream>>>(
        x, W_ih, W_hh, b_ih, b_hh, W_fc, b_fc, out);
}